// DFE_model_57423712748163
// MI455X (gfx1250) — compile-verified
//
#include <hip/hip_runtime.h>
#include <hip/hip_bf16.h>

// Problem constants (match reference).
#define MM   512      // batch rows m
#define JJ   512      // j dimension
#define KK   256      // k dimension
#define KH   128      // K/2 (X columns)
#define NNZ  16384

typedef __attribute__((ext_vector_type(2))) float v2f;
typedef __attribute__((ext_vector_type(8))) float v8f;

// ---------------------------------------------------------------------------
// Pass 1b: last-occurrence-wins dedup marking (reproduces JAX .set scatter
// with a deterministic choice among duplicates).
// ---------------------------------------------------------------------------
__global__ __launch_bounds__(256) void k_mark(const int* __restrict__ rows,
                                              const int* __restrict__ cols,
                                              unsigned* __restrict__ T) {
    int i = blockIdx.x * 256 + threadIdx.x;
    if (i < NNZ) {
        atomicMax(&T[rows[i] * KK + cols[i]], (unsigned)(i + 1));
    }
}

// ---------------------------------------------------------------------------
// Pass 1c: pack per-nnz SoA: coef = A_vals, colHalf = col>>1, vv = V[j,k],
// ww = W[j,k] (or 0 if this nnz lost the dedup), row = j.
// ---------------------------------------------------------------------------
__global__ __launch_bounds__(256) void k_pack(const float* __restrict__ vals,
                                              const int*   __restrict__ rows,
                                              const int*   __restrict__ cols,
                                              const float* __restrict__ V,
                                              const float* __restrict__ W,
                                              const unsigned* __restrict__ T,
                                              float* __restrict__ coefA,
                                              int*   __restrict__ colHalfA,
                                              float* __restrict__ vvA,
                                              float* __restrict__ wwA,
                                              int*   __restrict__ rowA) {
    int i = blockIdx.x * 256 + threadIdx.x;
    if (i >= NNZ) return;
    int r = rows[i];
    int c = cols[i];
    unsigned tv = T[r * KK + c];
    coefA[i]    = vals[i];
    colHalfA[i] = c >> 1;
    vvA[i]      = V[r * KK + c];
    wwA[i]      = (tv == (unsigned)(i + 1)) ? W[r * KK + c] : 0.0f;
    rowA[i]     = r;
}

// ---------------------------------------------------------------------------
// Pass 2: one block per m. Accumulate S[m, j] in LDS via ds_add_f32, then
// write Delta[m, j] = exp(-S[m, j]) to the workspace.
// ---------------------------------------------------------------------------
__global__ __launch_bounds__(256) void k_main(const float* __restrict__ X,
                                              const float* __restrict__ coefA,
                                              const int*   __restrict__ colHalfA,
                                              const float* __restrict__ vvA,
                                              const float* __restrict__ wwA,
                                              const int*   __restrict__ rowA,
                                              float* __restrict__ Delta) {
    __shared__ float Srow[JJ];
    __shared__ float Xrow[KH];
    const int m = blockIdx.x;
    const int t = threadIdx.x;

    Srow[t]       = 0.0f;
    Srow[t + 256] = 0.0f;
    if (t < KH) Xrow[t] = X[m * KH + t];
    __syncthreads();

    #pragma unroll 4
    for (int i = t; i < NNZ; i += 256) {
        float d = coefA[i] * Xrow[colHalfA[i]] - vvA[i];
        d = fmaxf(d, 0.0f);
        atomicAdd(&Srow[rowA[i]], wwA[i] * d * d);   // ds_add_f32
    }
    __syncthreads();

    Delta[m * JJ + t]       = expf(-Srow[t]);
    Delta[m * JJ + t + 256] = expf(-Srow[t + 256]);
}

// ---------------------------------------------------------------------------
// Pass 3: H = Delta(512x512) @ F(512) via V_WMMA_F32_16X16X4_F32 (exact f32).
// One wave (32 lanes) per 16-row tile of M. F is broadcast across all 16 N
// columns; the N=0 column of D is read back (lane 0 -> M 0..7, lane 16 ->
// M 8..15 per the CDNA5 16x16 f32 C/D layout).
// ---------------------------------------------------------------------------
__global__ __launch_bounds__(32) void k_wmma_matvec(const float* __restrict__ Delta,
                                                    const float* __restrict__ F,
                                                    float* __restrict__ H) {
    const int base = blockIdx.x * 16;          // 32 blocks cover M=512
    const int lane = threadIdx.x;              // wave32, EXEC all ones
    const int mrow = base + (lane & 15);
    const int k0   = (lane < 16) ? 0 : 2;      // A: VGPR0/1 = K {k0, k0+1}

    const float* Drow = Delta + mrow * JJ;
    v8f c = {};

    for (int kb = 0; kb < JJ; kb += 4) {
        // A tile (16x4): lane holds Delta[mrow][kb+k0], Delta[mrow][kb+k0+1]
        v2f a;
        const v2f* ap = reinterpret_cast<const v2f*>(Drow + kb + k0); // 8B aligned
        a = *ap;
        // B tile (4x16): row K value broadcast to every N column
        v2f b;
        b.x = F[kb + k0];
        b.y = F[kb + k0 + 1];
        // (neg_a, A, neg_b, B, c_mod, C, reuse_a, reuse_b)
        c = __builtin_amdgcn_wmma_f32_16x16x4_f32(
                false, a, false, b, (short)0, c, false, false);
    }

    // Extract column N=0 of the 16x16 D tile.
    if (lane == 0) {
        #pragma unroll
        for (int r = 0; r < 8; ++r) H[base + r] = c[r];
    }
    if (lane == 16) {
        #pragma unroll
        for (int r = 0; r < 8; ++r) H[base + 8 + r] = c[r];
    }
}

// ---------------------------------------------------------------------------
// Host-side launcher. Inputs (setup_inputs order):
//   0: X (512x128 f32), 1: A_vals (16384 f32), 2: V (512x256 f32),
//   3: W (512x256 f32), 4: Fvec (512 f32), 5: A_rows (16384 i32),
//   6: A_cols (16384 i32).  Output: H (512 f32).
// ---------------------------------------------------------------------------
extern "C" void kernel_launch(void* const* d_in, const int* in_sizes, int n_in,
                              void* d_out, int out_size, void* d_ws, size_t ws_size,
                              hipStream_t stream) {
    const float* X     = (const float*)d_in[0];
    const float* Avals = (const float*)d_in[1];
    const float* V     = (const float*)d_in[2];
    const float* W     = (const float*)d_in[3];
    const float* Fvec  = (const float*)d_in[4];
    const int*   Arows = (const int*)d_in[5];
    const int*   Acols = (const int*)d_in[6];
    float*       H     = (float*)d_out;

    // Workspace layout (bytes).
    char* ws = (char*)d_ws;
    unsigned* T     = (unsigned*)(ws);                        // 512 KB
    float* coefA    = (float*)(ws + 524288);                  //  64 KB
    int*   colHalfA = (int*)  (ws + 524288 + 65536);          //  64 KB
    float* vvA      = (float*)(ws + 524288 + 2 * 65536);      //  64 KB
    float* wwA      = (float*)(ws + 524288 + 3 * 65536);      //  64 KB
    int*   rowA     = (int*)  (ws + 524288 + 4 * 65536);      //  64 KB
    float* Delta    = (float*)(ws + 524288 + 5 * 65536);      //   1 MB

    // 1a: zero dedup table (graph-capture-safe async memset).
    hipMemsetAsync(T, 0, (size_t)JJ * KK * sizeof(unsigned), stream);

    // 1b: last-occurrence marking.
    k_mark<<<NNZ / 256, 256, 0, stream>>>(Arows, Acols, T);

    // 1c: per-nnz SoA pack with dedup + V/W gather.
    k_pack<<<NNZ / 256, 256, 0, stream>>>(Avals, Arows, Acols, V, W, T,
                                          coefA, colHalfA, vvA, wwA, rowA);

    // 2: scatter-accumulate S per m in LDS, write Delta = exp(-S).
    k_main<<<MM, 256, 0, stream>>>(X, coefA, colHalfA, vvA, wwA, rowA, Delta);

    // 3: H = Delta @ F on the matrix pipe (f32 WMMA, exact).
    k_wmma_matvec<<<MM / 16, 32, 0, stream>>>(Delta, Fvec, H);
}